// GM_36464272343072
// MI455X (gfx1250) — compile-verified
//
#include <hip/hip_runtime.h>
#include <hip/hip_bf16.h>
#include <math.h>

// ---------------- problem constants (match reference) ----------------
#define N_PROP   50000
#define N_CLASS  81
#define CM1      80           // classes 1..80 (skip background)
#define IMG_W    1216.0f
#define IMG_H    800.0f
#define EPS_BB   1e-5f
#define SCORE_TH 0.05f
#define NMS_TH   0.3f
#define TOPK     1000
#define KPAD     1008         // TOPK padded to 63 tiles of 16
#define NPAD     65536        // pow2 pad of N_PROP for bitonic sort
#define NTILE    63           // KPAD / 16

typedef float v2f __attribute__((ext_vector_type(2)));
typedef float v8f __attribute__((ext_vector_type(8)));

// ---- monotone u64 key: high 32 = descending-score map, low 32 = index ----
// ascending sort of key  ==  score descending, index ascending (JAX stable top_k)
__device__ __forceinline__ unsigned int mono_desc(float s) {
    unsigned int b = __float_as_uint(s);
    unsigned int u = (b & 0x80000000u) ? ~b : (b | 0x80000000u); // ascending map
    return ~u;                                                   // descending
}
__device__ __forceinline__ unsigned long long make_key(float p, unsigned int n) {
    float s = (p > SCORE_TH) ? p : -1.0f;   // reference: thresholded else -1
    return ((unsigned long long)mono_desc(s) << 32) | (unsigned long long)n;
}

// =====================================================================
// Stage A: softmax over 81 classes per proposal (one wave32 / proposal),
//          emit sortable keys transposed to class-major [CM1][NPAD].
// =====================================================================
__global__ __launch_bounds__(256) void softmax_key_kernel(
        const float* __restrict__ pred_scores, unsigned long long* __restrict__ keys) {
    int wave = blockIdx.x * 8 + (threadIdx.x >> 5);
    int lane = threadIdx.x & 31;
    if (wave >= N_PROP) return;
    const float* row = pred_scores + (size_t)wave * N_CLASS;

    float x0 = row[lane];                                  // 0..31
    float x1 = row[lane + 32];                             // 32..63
    bool  h2 = (lane + 64) < N_CLASS;                      // 64..80
    float x2 = h2 ? row[lane + 64] : -3.0e38f;

    float m = fmaxf(fmaxf(x0, x1), x2);
    #pragma unroll
    for (int off = 16; off > 0; off >>= 1) m = fmaxf(m, __shfl_xor(m, off, 32));
    float e0 = __expf(x0 - m), e1 = __expf(x1 - m), e2 = h2 ? __expf(x2 - m) : 0.0f;
    float sum = e0 + e1 + e2;
    #pragma unroll
    for (int off = 16; off > 0; off >>= 1) sum += __shfl_xor(sum, off, 32);
    float inv = 1.0f / sum;

    unsigned int n = (unsigned int)wave;
    if (lane >= 1)
        keys[(size_t)(lane - 1) * NPAD + n] = make_key(e0 * inv, n);   // classes 1..31
    keys[(size_t)(lane + 31) * NPAD + n] = make_key(e1 * inv, n);       // classes 32..63
    if (h2)
        keys[(size_t)(lane + 63) * NPAD + n] = make_key(e2 * inv, n);   // classes 64..80
}

// pad keys[ N_PROP .. NPAD ) with +inf key (sorts last)
__global__ void pad_keys_kernel(unsigned long long* __restrict__ keys) {
    int tid = blockIdx.x * blockDim.x + threadIdx.x;
    const int per = NPAD - N_PROP;
    if (tid >= CM1 * per) return;
    int c = tid / per, n = N_PROP + tid % per;
    keys[(size_t)c * NPAD + n] = ~0ull;
}

// =====================================================================
// Stage B: bitonic sort (ascending u64 key) per class, 80 classes parallel
// =====================================================================
__global__ __launch_bounds__(256) void bitonic_pass_kernel(
        unsigned long long* __restrict__ keys, int j, int k) {
    int tid = blockIdx.x * blockDim.x + threadIdx.x;
    if (tid >= CM1 * NPAD) return;
    int c = tid >> 16;
    int i = tid & (NPAD - 1);
    int ixj = i ^ j;
    if (ixj > i) {
        unsigned long long* base = keys + ((size_t)c << 16);
        unsigned long long va = base[i], vb = base[ixj];
        bool up = ((i & k) == 0);
        if ((va > vb) == up) { base[i] = vb; base[ixj] = va; }
    }
}

// =====================================================================
// Stage C: decode bbox for the top-TOPK (per class) selected proposals
// =====================================================================
__global__ __launch_bounds__(256) void decode_select_kernel(
        const unsigned long long* __restrict__ keys,
        const float* __restrict__ proposals, const float* __restrict__ pred_bbox,
        const float* __restrict__ mean, const float* __restrict__ stdv,
        float4* __restrict__ boxesSel, float* __restrict__ scoresSel) {
    int gid = blockIdx.x * blockDim.x + threadIdx.x;
    if (gid >= CM1 * KPAD) return;
    int c = gid / KPAD, k = gid % KPAD;

    float4 bx = {0.f, 0.f, 0.f, 0.f};
    float  s  = 0.f;
    if (k < TOPK) {
        unsigned long long key = keys[(size_t)c * NPAD + k];
        unsigned int idx = (unsigned int)(key & 0xFFFFFFFFu);
        unsigned int u   = ~((unsigned int)(key >> 32));
        unsigned int fb  = (u & 0x80000000u) ? (u & 0x7FFFFFFFu) : ~u;
        if (idx < N_PROP) {
            s = __uint_as_float(fb);                 // top_s (== -1 for invalid fills)
            int cls = c + 1;
            float px1 = proposals[idx * 4 + 0], py1 = proposals[idx * 4 + 1];
            float px2 = proposals[idx * 4 + 2], py2 = proposals[idx * 4 + 3];
            float w  = px2 - px1 + EPS_BB, h = py2 - py1 + EPS_BB;
            float cx = px1 + 0.5f * w,     cy = py1 + 0.5f * h;
            size_t db = (size_t)idx * (4 * N_CLASS) + (size_t)cls * 4;
            float d0 = pred_bbox[db + 0] * stdv[cls * 4 + 0] + mean[cls * 4 + 0];
            float d1 = pred_bbox[db + 1] * stdv[cls * 4 + 1] + mean[cls * 4 + 1];
            float d2 = pred_bbox[db + 2] * stdv[cls * 4 + 2] + mean[cls * 4 + 2];
            float d3 = pred_bbox[db + 3] * stdv[cls * 4 + 3] + mean[cls * 4 + 3];
            float pcx = d0 * w + cx, pcy = d1 * h + cy;
            float pw = __expf(d2) * w, ph = __expf(d3) * h;
            bx.x = fminf(fmaxf(pcx - 0.5f * pw, 0.f), IMG_W);
            bx.y = fminf(fmaxf(pcy - 0.5f * ph, 0.f), IMG_H);
            bx.z = fminf(fmaxf(pcx + 0.5f * pw, 0.f), IMG_W);
            bx.w = fminf(fmaxf(pcy + 0.5f * ph, 0.f), IMG_H);
        }
    }
    boxesSel[(size_t)c * KPAD + k]  = bx;
    scoresSel[(size_t)c * KPAD + k] = s;
}

// =====================================================================
// Stage D1: suppression bit-matrix via 16x16 tiles, one wave32 per tile.
// Union base (area_i + area_j) for the whole tile is produced with one
// V_WMMA_F32_16X16X4_F32:  D = A(16x4) x B(4x16),
//   A[:,0]=area_row, A[:,1]=1, B[0,:]=1, B[1,:]=area_col  (K=2,3 zero).
// ISA VGPR layouts (CDNA5 ISA):
//   A/B: lanes 0-15 hold K=0,1 ; lanes 16-31 hold K=2,3 (zeros here)
//   D:   vgpr k, lanes 0-15 -> M=k, N=lane ; lanes 16-31 -> M=k+8, N=lane-16
// ballot_w32(pred) therefore yields bits[15:0]=row k, bits[31:16]=row k+8.
// The IoU threshold is evaluated division-free:
//   inter/max(uni,eps) > TH  <=>  inter > TH*max(uni,eps)   (denominator > 0)
// =====================================================================
__global__ __launch_bounds__(32) void pair_mask_kernel(
        const float4* __restrict__ boxesSel, unsigned short* __restrict__ maskRows) {
    int q = blockIdx.x;          // col tile
    int r = blockIdx.y;          // row tile
    int c = blockIdx.z;          // class
    if (q < r) return;           // only j>i is ever consumed by the scan

    int lane = threadIdx.x;
    const float4* B = boxesSel + (size_t)c * KPAD;

    float4 cb = B[q * 16 + (lane & 15)];                   // this lane's column box
    float  areaC = (cb.z - cb.x) * (cb.w - cb.y);
    float4 rb = B[r * 16 + (lane & 15)];
    float  areaR = (rb.z - rb.x) * (rb.w - rb.y);

    bool lo = (lane < 16);
    v2f a, b;
    a.x = lo ? areaR : 0.0f;  a.y = lo ? 1.0f : 0.0f;      // A[:,{0,1}] / A[:,{2,3}]=0
    b.x = lo ? 1.0f  : 0.0f;  b.y = lo ? areaC : 0.0f;     // B[{0,1},:] / B[{2,3},:]=0

    v8f acc = {};
    acc = __builtin_amdgcn_wmma_f32_16x16x4_f32(
        /*neg_a=*/false, a, /*neg_b=*/false, b,
        /*c_mod=*/(short)0, acc, /*reuse_a=*/false, /*reuse_b=*/false);

    int base = (lane >> 4) * 8;                            // 0 for lanes 0-15, 8 for 16-31
    #pragma unroll
    for (int k = 0; k < 8; ++k) {
        float4 pb = B[r * 16 + base + k];                  // row box for this D element
        float x1 = fmaxf(pb.x, cb.x), y1 = fmaxf(pb.y, cb.y);
        float x2 = fminf(pb.z, cb.z), y2 = fminf(pb.w, cb.w);
        float inter = fmaxf(x2 - x1, 0.0f) * fmaxf(y2 - y1, 0.0f);
        float uni   = acc[k] - inter;                      // area_i + area_j - inter
        // division-free IoU threshold (denominator strictly positive)
        bool  sup   = inter > NMS_TH * fmaxf(uni, 1e-10f);
        unsigned int bal = __builtin_amdgcn_ballot_w32(sup);
        size_t rowLo = ((size_t)c * KPAD + r * 16 + k) * 64 + q;
        size_t rowHi = ((size_t)c * KPAD + r * 16 + k + 8) * 64 + q;
        if (lane == 0)  maskRows[rowLo] = (unsigned short)(bal & 0xFFFFu);
        if (lane == 16) maskRows[rowHi] = (unsigned short)(bal >> 16);
    }
}

// =====================================================================
// Stage D2: sequential greedy scan over the bitmask (== reference fori_loop)
//           + final [TOPK,5] output, one 64-thread block per class.
// =====================================================================
__global__ __launch_bounds__(64) void nms_scan_kernel(
        const float4* __restrict__ boxesSel, const float* __restrict__ scoresSel,
        const unsigned short* __restrict__ maskRows, float* __restrict__ out) {
    int c = blockIdx.x;
    int t = threadIdx.x;                  // word owner: cols 16t..16t+15
    __shared__ unsigned short keepW[64];

    unsigned int w = 0;
    #pragma unroll
    for (int bit = 0; bit < 16; ++bit) {
        int j = t * 16 + bit;
        if (j < TOPK && scoresSel[(size_t)c * KPAD + j] > SCORE_TH) w |= (1u << bit);
    }
    keepW[t] = (unsigned short)w;
    __syncthreads();

    const unsigned short* mrow = maskRows + (size_t)c * KPAD * 64;
    for (int i = 0; i < TOPK; ++i) {
        int pw = i >> 4, pb = i & 15;
        bool ki = (keepW[pw] >> pb) & 1;        // bit i never cleared at step i
        if (ki) {
            unsigned int m = mrow[(size_t)i * 64 + t];
            if (t < pw)       m = 0;                                   // j <= i
            else if (t == pw) m &= ~((1u << (pb + 1)) - 1u);           // keep j > i
            keepW[t] = (unsigned short)(keepW[t] & ~m);
        }
        __syncthreads();
    }

    for (int k = t; k < TOPK; k += 64) {
        bool kp = (keepW[k >> 4] >> (k & 15)) & 1;
        float4 bx = boxesSel[(size_t)c * KPAD + k];
        float  s  = scoresSel[(size_t)c * KPAD + k];
        float* o = out + ((size_t)c * TOPK + k) * 5;
        o[0] = kp ? bx.x : 0.f;  o[1] = kp ? bx.y : 0.f;
        o[2] = kp ? bx.z : 0.f;  o[3] = kp ? bx.w : 0.f;
        o[4] = kp ? s    : 0.f;
    }
}

// =====================================================================
// host-side orchestration (graph-capture safe: launches only, on `stream`)
// =====================================================================
extern "C" void kernel_launch(void* const* d_in, const int* in_sizes, int n_in,
                              void* d_out, int out_size, void* d_ws, size_t ws_size,
                              hipStream_t stream) {
    const float* proposals   = (const float*)d_in[0];   // [N,4]
    const float* pred_bbox   = (const float*)d_in[1];   // [N,4*81]
    const float* pred_scores = (const float*)d_in[2];   // [N,81]
    const float* mean        = (const float*)d_in[3];   // [324]
    const float* stdv        = (const float*)d_in[4];   // [324]
    float* out = (float*)d_out;                         // [80,1000,5]

    // workspace layout (bytes):
    //   keys      : 80 * 65536 * 8   = 41,943,040
    //   boxesSel  : 80 * 1008  * 16  =  1,290,240
    //   scoresSel : 80 * 1008  * 4   =    322,560
    //   maskRows  : 80 * 1008 * 64*2 = 10,321,920      (total ~53.9 MB)
    char* ws = (char*)d_ws;
    unsigned long long* keys = (unsigned long long*)ws;
    float4* boxesSel   = (float4*)(ws + 41943040);
    float*  scoresSel  = (float*)(ws + 41943040 + 1290240);
    unsigned short* maskRows = (unsigned short*)(ws + 41943040 + 1290240 + 322560);

    // A: softmax + sortable keys (one wave32 per proposal)
    softmax_key_kernel<<<N_PROP / 8, 256, 0, stream>>>(pred_scores, keys);
    {
        int total = CM1 * (NPAD - N_PROP);
        pad_keys_kernel<<<(total + 255) / 256, 256, 0, stream>>>(keys);
    }

    // B: bitonic sort per class (ascending key == descending score, stable ties)
    {
        int total = CM1 * NPAD;
        int blocks = (total + 255) / 256;
        for (int k = 2; k <= NPAD; k <<= 1)
            for (int j = k >> 1; j > 0; j >>= 1)
                bitonic_pass_kernel<<<blocks, 256, 0, stream>>>(keys, j, k);
    }

    // C: decode top-1000 boxes per class (pad to 1008 with zero boxes)
    {
        int total = CM1 * KPAD;
        decode_select_kernel<<<(total + 255) / 256, 256, 0, stream>>>(
            keys, proposals, pred_bbox, mean, stdv, boxesSel, scoresSel);
    }

    // D1: WMMA-tiled suppression bit-matrix (one wave32 per 16x16 IoU tile)
    pair_mask_kernel<<<dim3(NTILE, NTILE, CM1), 32, 0, stream>>>(boxesSel, maskRows);

    // D2: greedy bitmask scan + output, one block per class
    nms_scan_kernel<<<CM1, 64, 0, stream>>>(boxesSel, scoresSel, maskRows, out);
}